// OnlineTripletLoss_24739011625759
// MI455X (gfx1250) — compile-verified
//
#include <hip/hip_runtime.h>
#include <hip/hip_bf16.h>

// ---------------------------------------------------------------------------
// Online triplet loss with batch-hard mining, fused fp32-WMMA Gram matrix.
//   N=8192 points, D=256 dims, 64 classes, margin 0.2
// dist[i,j] = -2*(E@E^T)[i,j] + sq[i] + sq[j]; hardest pos = argmax over same
// class, hardest neg = argmin over different class; loss = mean relu(ap-an+m).
// The 8192x8192 distance matrix is never materialized: mining is fused into
// the GEMM epilogue. GEMM runs on V_WMMA_F32_16X16X4_F32 (full fp32).
// B tiles are staged once per workgroup in double-buffered LDS and shared by
// all 8 waves (8x cut in L2 traffic vs per-wave streaming). DS reads are
// software-pipelined one fragment-group ahead of the WMMA chain.
// ---------------------------------------------------------------------------

typedef float v2f __attribute__((ext_vector_type(2)));
typedef float v8f __attribute__((ext_vector_type(8)));

#define N_PTS   8192
#define DIMS    256
#define MARGIN  0.2f
#define CSPLIT  8                       // column split for parallelism
#define COLS_PER_SPLIT (N_PTS / CSPLIT) // 1024
#define WAVES_PER_WG 8
#define ROWS_PER_WG  (WAVES_PER_WG * 16) // 128
#define LDS_PITCH 260                   // 256 + 4 pad -> conflict-free b64 reads

// ---------------------------------------------------------------------------
// Kernel 1: squared norms, one wave32 per row (256 floats -> 8/lane).
// ---------------------------------------------------------------------------
__global__ void sqnorm_kernel(const float* __restrict__ E, float* __restrict__ sq) {
    int wave = (int)((blockIdx.x * blockDim.x + threadIdx.x) >> 5);
    int lane = (int)(threadIdx.x & 31);
    if (wave >= N_PTS) return;
    const float4* row = (const float4*)(E + (size_t)wave * DIMS);
    float4 a = row[lane];        // cols lane*4 .. +3
    float4 b = row[lane + 32];   // cols 128 + lane*4 .. +3
    float s = a.x*a.x + a.y*a.y + a.z*a.z + a.w*a.w
            + b.x*b.x + b.y*b.y + b.z*b.z + b.w*b.w;
    #pragma unroll
    for (int m = 16; m >= 1; m >>= 1) s += __shfl_xor(s, m, 32);
    if (lane == 0) sq[wave] = s;
}

// ---------------------------------------------------------------------------
// Kernel 2: fused WMMA GEMM + masked argmax/argmin mining.
// grid = (N/128, CSPLIT), block = 256 (8 waves). Wave w owns 16 rows,
// block.y selects a 1024-column slice. A-strip cached in registers
// (fully unrolled -> register-promoted). B tile (16 cols x 256 K = 16KB)
// staged in double-buffered LDS, shared by all 8 waves.
//
// Fragment layout (32-bit 16x16x4, wave32):
//   A: lane L holds row M = L%16, K pair { 4k+2*(L/16), +1 }  -> v2f
//   B: lane L holds col N = L%16, same K pair (B of E@E^T is rows of E)
//   C: lane L, elem v holds M = v + 8*(L/16), N = L%16
// LDS tile layout: bt[buf][col][k] with pitch 260 floats; a ds_load_b64 at
// float offset 260*lo + 8p + 2*hi maps the 32 lanes onto 64 distinct banks.
// ---------------------------------------------------------------------------
__global__ __launch_bounds__(256, 1)
void mine_kernel(const float* __restrict__ E, const int* __restrict__ cls,
                 const float* __restrict__ sq,
                 float* __restrict__ pPosV, int* __restrict__ pPosI,
                 float* __restrict__ pNegV, int* __restrict__ pNegI) {
    __shared__ float bt[2][16][LDS_PITCH];

    const int tid  = (int)threadIdx.x;
    const int lane = tid & 31;
    const int wave = tid >> 5;
    const int hi   = lane >> 4;     // 0 or 1
    const int lo   = lane & 15;
    const int rowBase = (int)blockIdx.x * ROWS_PER_WG + wave * 16;
    const int colBeg  = (int)blockIdx.y * COLS_PER_SPLIT;
    const int NT      = COLS_PER_SPLIT / 16;   // 64 column tiles

    // staging coordinates: 256 threads cover 16 rows x 4 float4 chunks
    const int srow = tid >> 4;      // 0..15: tile row (= column of dist)
    const int sqf  = tid & 15;      // 0..15: float4 group within the row

    // ---- cache A fragments for this wave's 16 rows (64 x v2f = 128 VGPRs).
    const float* arow = E + (size_t)(rowBase + lo) * DIMS + 2 * hi;
    v2f aF[DIMS / 4];
    #pragma unroll
    for (int kk = 0; kk < DIMS / 4; ++kk)
        aF[kk] = *(const v2f*)(arow + kk * 4);

    // ---- per-lane row-side stats for accumulator rows M = v + 8*hi
    float sqRow[8]; int clsRow[8];
    #pragma unroll
    for (int v = 0; v < 8; ++v) {
        int r = rowBase + 8 * hi + v;
        sqRow[v]  = sq[r];
        clsRow[v] = cls[r];
    }

    float bestPV[8], bestNV[8]; int bestPI[8], bestNI[8];
    #pragma unroll
    for (int v = 0; v < 8; ++v) {
        bestPV[v] = -1e30f; bestPI[v] = 0;
        bestNV[v] =  1e30f; bestNI[v] = 0;
    }

    // ---- prologue: stage tile 0 into buffer 0
    {
        const float* g = E + (size_t)(colBeg + srow) * DIMS + 4 * sqf;
        #pragma unroll
        for (int j = 0; j < 4; ++j) {
            float4 x = *(const float4*)(g + 64 * j);
            *(float4*)(&bt[0][srow][4 * sqf + 64 * j]) = x;
        }
    }
    __syncthreads();

    for (int t = 0; t < NT; ++t) {
        // ---- issue next tile's global loads early (latency hidden by WMMAs)
        float4 stg[4];
        if (t + 1 < NT) {
            const float* g = E + (size_t)(colBeg + (t + 1) * 16 + srow) * DIMS + 4 * sqf;
            #pragma unroll
            for (int j = 0; j < 4; ++j)
                stg[j] = *(const float4*)(g + 64 * j);
        }

        const int col  = colBeg + t * 16 + lo;      // this lane's N column
        const float sqC  = sq[col];
        const int   clsC = cls[col];

        // ---- consume current tile from LDS: 64 chained WMMAs, 2 acc chains.
        // Software pipeline: fragment group g+1 (4 x ds_load_2addr_b64) is
        // issued before the 8 WMMAs of group g, so dscnt waits overlap math.
        const float* lb = &bt[t & 1][0][0] + lo * LDS_PITCH + 2 * hi;
        v8f acc0 = {};
        v8f acc1 = {};
        v2f b[8], bn[8];
        #pragma unroll
        for (int q = 0; q < 4; ++q) {
            b[2 * q]     = *(const v2f*)(lb + q * 8);
            b[2 * q + 1] = *(const v2f*)(lb + q * 8 + 4);
        }
        #pragma unroll
        for (int g = 0; g < 8; ++g) {
            if (g < 7) {
                #pragma unroll
                for (int q = 0; q < 4; ++q) {
                    bn[2 * q]     = *(const v2f*)(lb + (g + 1) * 32 + q * 8);
                    bn[2 * q + 1] = *(const v2f*)(lb + (g + 1) * 32 + q * 8 + 4);
                }
            }
            #pragma unroll
            for (int q = 0; q < 4; ++q) {
                acc0 = __builtin_amdgcn_wmma_f32_16x16x4_f32(
                        false, aF[8 * g + 2 * q],     false, b[2 * q],     (short)0, acc0, false, false);
                acc1 = __builtin_amdgcn_wmma_f32_16x16x4_f32(
                        false, aF[8 * g + 2 * q + 1], false, b[2 * q + 1], (short)0, acc1, false, false);
            }
            #pragma unroll
            for (int q = 0; q < 8; ++q) b[q] = bn[q];   // register rename (SSA)
        }
        v8f acc = acc0 + acc1;

        // ---- epilogue: dist + masked running best
        #pragma unroll
        for (int v = 0; v < 8; ++v) {
            float d  = fmaf(-2.0f, acc[v], sqRow[v] + sqC);
            bool same = (clsRow[v] == clsC);
            float pc = same ? d : -1e30f;           // positive candidate
            float nc = same ? 1e30f : d;            // negative candidate
            if (pc > bestPV[v]) { bestPV[v] = pc; bestPI[v] = col; }
            if (nc < bestNV[v]) { bestNV[v] = nc; bestNI[v] = col; }
        }

        // ---- drain staged registers into the other LDS buffer
        if (t + 1 < NT) {
            #pragma unroll
            for (int j = 0; j < 4; ++j)
                *(float4*)(&bt[(t + 1) & 1][srow][4 * sqf + 64 * j]) = stg[j];
        }
        __syncthreads();
    }

    // ---- reduce across the 16 lanes of each N-group (xor keeps hi fixed)
    #pragma unroll
    for (int v = 0; v < 8; ++v) {
        float pv = bestPV[v]; int pi = bestPI[v];
        float nv = bestNV[v]; int ni = bestNI[v];
        #pragma unroll
        for (int m = 1; m < 16; m <<= 1) {
            float opv = __shfl_xor(pv, m, 32); int opi = __shfl_xor(pi, m, 32);
            float onv = __shfl_xor(nv, m, 32); int oni = __shfl_xor(ni, m, 32);
            if (opv > pv || (opv == pv && opi < pi)) { pv = opv; pi = opi; }
            if (onv < nv || (onv == nv && oni < ni)) { nv = onv; ni = oni; }
        }
        if (lo == 0) {
            int r = rowBase + 8 * hi + v;
            int o = r * CSPLIT + (int)blockIdx.y;
            pPosV[o] = pv; pPosI[o] = pi;
            pNegV[o] = nv; pNegI[o] = ni;
        }
    }
}

// ---------------------------------------------------------------------------
// Kernel 3: merge the CSPLIT partials per row (first-index tie-break, split
// order == ascending column order), emit triplets + per-row loss.
// ---------------------------------------------------------------------------
__global__ void merge_kernel(const float* __restrict__ pPosV, const int* __restrict__ pPosI,
                             const float* __restrict__ pNegV, const int* __restrict__ pNegI,
                             float* __restrict__ out, float* __restrict__ loss) {
    int r = (int)(blockIdx.x * blockDim.x + threadIdx.x);
    if (r >= N_PTS) return;
    float pv = -1e30f; int pi = 0;
    float nv =  1e30f; int ni = 0;
    #pragma unroll
    for (int s = 0; s < CSPLIT; ++s) {
        float v = pPosV[r * CSPLIT + s]; int i = pPosI[r * CSPLIT + s];
        if (v > pv || (v == pv && i < pi)) { pv = v; pi = i; }
        float w = pNegV[r * CSPLIT + s]; int j = pNegI[r * CSPLIT + s];
        if (w < nv || (w == nv && j < ni)) { nv = w; ni = j; }
    }
    float l = pv - nv + MARGIN;                 // ap - an + margin
    loss[r] = l > 0.0f ? l : 0.0f;
    out[1 + 3 * r + 0] = (float)r;              // anchor
    out[1 + 3 * r + 1] = (float)pi;             // hardest positive
    out[1 + 3 * r + 2] = (float)ni;             // hardest negative
}

// ---------------------------------------------------------------------------
// Kernel 4: deterministic single-block mean reduction.
// ---------------------------------------------------------------------------
__global__ void mean_kernel(const float* __restrict__ loss, float* __restrict__ out) {
    __shared__ float sm[256];
    float s = 0.0f;
    for (int i = (int)threadIdx.x; i < N_PTS; i += 256) s += loss[i];
    sm[threadIdx.x] = s;
    __syncthreads();
    for (int w = 128; w > 0; w >>= 1) {
        if ((int)threadIdx.x < w) sm[threadIdx.x] += sm[threadIdx.x + w];
        __syncthreads();
    }
    if (threadIdx.x == 0) out[0] = sm[0] * (1.0f / N_PTS);
}

// ---------------------------------------------------------------------------
extern "C" void kernel_launch(void* const* d_in, const int* in_sizes, int n_in,
                              void* d_out, int out_size, void* d_ws, size_t ws_size,
                              hipStream_t stream) {
    (void)in_sizes; (void)n_in; (void)out_size; (void)ws_size;
    const float* E   = (const float*)d_in[0];   // (8192, 256) fp32
    const int*   tgt = (const int*)d_in[1];     // (8192,) class ids
    float* out = (float*)d_out;                 // [mean_loss, (a,p,n)*8192]

    char* ws = (char*)d_ws;
    float* sq    = (float*)ws;                      ws += (size_t)N_PTS * 4;
    float* pPosV = (float*)ws;                      ws += (size_t)N_PTS * CSPLIT * 4;
    int*   pPosI = (int*)ws;                        ws += (size_t)N_PTS * CSPLIT * 4;
    float* pNegV = (float*)ws;                      ws += (size_t)N_PTS * CSPLIT * 4;
    int*   pNegI = (int*)ws;                        ws += (size_t)N_PTS * CSPLIT * 4;
    float* loss  = (float*)ws;

    // 1) squared norms: one wave per row
    sqnorm_kernel<<<dim3(N_PTS / 8), dim3(256), 0, stream>>>(E, sq);

    // 2) fused WMMA mining: 64 row-blocks x 8 column splits, 256 thr (8 waves)
    mine_kernel<<<dim3(N_PTS / ROWS_PER_WG, CSPLIT), dim3(256), 0, stream>>>(
        E, tgt, sq, pPosV, pPosI, pNegV, pNegI);

    // 3) merge partials, emit triplets + per-row loss
    merge_kernel<<<dim3(N_PTS / 256), dim3(256), 0, stream>>>(
        pPosV, pPosI, pNegV, pNegI, out, loss);

    // 4) deterministic mean
    mean_kernel<<<dim3(1), dim3(256), 0, stream>>>(loss, out);
}